// Attn_60705067761886
// MI455X (gfx1250) — compile-verified
//
#include <hip/hip_runtime.h>

#define H 1024
#define S 32768

// energies kernel tiling
#define ROWS   128          // rows per block (8 waves x 16)
#define CK     64           // K columns per stage
#define STAGES (H / CK)     // 16
#define PITCH  68           // dwords per staged row (16B aligned, bank-conflict-free)
#define BUFSZ  (ROWS * PITCH)

typedef __attribute__((ext_vector_type(2))) float v2f;
typedef __attribute__((ext_vector_type(8))) float v8f;
typedef __attribute__((ext_vector_type(4))) int   v4i;

typedef __attribute__((address_space(1))) v4i* gv4i_p;   // global int4*
typedef __attribute__((address_space(3))) v4i* lv4i_p;   // LDS int4*

// ---------------------------------------------------------------------------
// async global->LDS B128 copy (gfx1250 ASYNCcnt path)
// ---------------------------------------------------------------------------
__device__ __forceinline__ void async_b128(const float* g, float* l) {
#if defined(__gfx1250__) && __has_builtin(__builtin_amdgcn_global_load_async_to_lds_b128)
  __builtin_amdgcn_global_load_async_to_lds_b128((gv4i_p)g, (lv4i_p)l, 0, 0);
#else
  unsigned loff = (unsigned)(unsigned long long)l;  // LDS aperture low bits = LDS offset
  asm volatile("global_load_async_to_lds_b128 %0, %1, off"
               :: "v"(loff), "v"(g) : "memory");
#endif
}

template <int N>
__device__ __forceinline__ void wait_async() {
#if defined(__gfx1250__) && __has_builtin(__builtin_amdgcn_s_wait_asynccnt)
  __builtin_amdgcn_s_wait_asynccnt(N);
#else
  asm volatile("s_wait_asynccnt %0" :: "i"(N) : "memory");
#endif
}

// ---------------------------------------------------------------------------
// Kernel 1: ws[0..H-1] = v = W^T h   ;   ws[H] = c = b . h
// ---------------------------------------------------------------------------
__global__ void attn_projvec_kernel(const float* __restrict__ W,
                                    const float* __restrict__ h,
                                    const float* __restrict__ b,
                                    float* __restrict__ ws) {
  if (blockIdx.x < 4) {
    const int j = blockIdx.x * 256 + threadIdx.x;
    float acc = 0.0f;
#pragma unroll 4
    for (int i = 0; i < H; ++i) acc = fmaf(W[(size_t)i * H + j], h[i], acc);
    ws[j] = acc;
  } else {
    __shared__ float red[256];
    float acc = 0.0f;
    for (int i = threadIdx.x; i < H; i += 256) acc = fmaf(b[i], h[i], acc);
    red[threadIdx.x] = acc;
    __syncthreads();
    for (int s = 128; s > 0; s >>= 1) {
      if (threadIdx.x < s) red[threadIdx.x] += red[threadIdx.x + s];
      __syncthreads();
    }
    if (threadIdx.x == 0) ws[H] = red[0];
  }
}

// ---------------------------------------------------------------------------
// Kernel 2: energies[s] = enc[s].v + c  via V_WMMA_F32_16X16X4_F32 with
// double-buffered async global->LDS staging of enc tiles.
// ---------------------------------------------------------------------------
__global__ void __launch_bounds__(256)
attn_energies_kernel(const float* __restrict__ enc,
                     const float* __restrict__ ws,
                     float* __restrict__ energies) {
  __shared__ float smem[2 * BUFSZ + H];   // two stage buffers + v
  float* vsh = smem + 2 * BUFSZ;

  const int tid = threadIdx.x;
  const float* gbase = enc + (size_t)blockIdx.x * ROWS * H;

  // stage v into LDS (visible after first barrier below)
  for (int i = tid; i < H; i += 256) vsh[i] = ws[i];
  const float cbh = ws[H];

  // ---- async staging helper: stage s covers columns [s*CK, s*CK+CK) ----
  // 2048 16B-chunks per stage (128 rows x 16 chunks), 8 per thread.
  auto stage_copy = [&](int s, float* buf) {
#pragma unroll
    for (int i = 0; i < 8; ++i) {
      const int p = tid + 256 * i;
      const int r = p >> 4;              // row 0..127
      const int q = (p & 15) << 2;       // float offset 0,4,..,60
      async_b128(gbase + (size_t)r * H + s * CK + q, buf + r * PITCH + q);
    }
  };

  const int lane = tid & 31;
  const int wave = tid >> 5;
  const int m    = lane & 15;
  const int kh   = (lane >> 4) << 1;     // 0 or 2: K-pair of this lane half
  const int rb   = wave * 16 + m;        // staged row this lane reads

  v8f acc = {cbh, cbh, cbh, cbh, cbh, cbh, cbh, cbh};

  stage_copy(0, smem);                   // prologue -> buffer 0

  for (int s = 0; s < STAGES; ++s) {
    float* cur = smem + (s & 1) * BUFSZ;
    if (s + 1 < STAGES) {
      stage_copy(s + 1, smem + ((s + 1) & 1) * BUFSZ);
      wait_async<8>();                   // in-order: stage s's 8 copies landed
    } else {
      wait_async<0>();
    }
    __syncthreads();                     // all waves' copies of stage s visible

    const float* arow = cur + rb * PITCH + kh;
    const float* vrow = vsh + s * CK + kh;
#pragma unroll
    for (int kk = 0; kk < CK; kk += 4) {
      v2f a  = *(const v2f*)(arow + kk); // ds_load_b64, conflict-free (pitch 68)
      v2f bb = *(const v2f*)(vrow + kk); // ds_load_b64, broadcast
      acc = __builtin_amdgcn_wmma_f32_16x16x4_f32(
          false, a, false, bb, (short)0, acc, false, false);
    }
    __syncthreads();                     // buffer reusable for stage s+2
  }

  // All D columns identical. Lane 0 holds rows +0..7 in acc[0..7]; lane 16
  // holds rows +8..15.
  if ((lane & 15) == 0) {
    const int rbase = blockIdx.x * ROWS + wave * 16 + (lane >> 4) * 8;
#pragma unroll
    for (int r = 0; r < 8; ++r) energies[rbase + r] = acc[r];
  }
}

// ---------------------------------------------------------------------------
// Kernel 3: numerically-stable softmax over S values, single 1024-thread
// block, deterministic fixed-order reductions.
// ---------------------------------------------------------------------------
__global__ void attn_softmax_kernel(const float* __restrict__ energies,
                                    float* __restrict__ out) {
  __shared__ float red[1024];
  const int tid = threadIdx.x;

  float mx = -3.402823466e38f;
  for (int i = tid; i < S; i += 1024) mx = fmaxf(mx, energies[i]);
  red[tid] = mx;
  __syncthreads();
  for (int s = 512; s > 0; s >>= 1) {
    if (tid < s) red[tid] = fmaxf(red[tid], red[tid + s]);
    __syncthreads();
  }
  mx = red[0];
  __syncthreads();

  float sum = 0.0f;
  for (int i = tid; i < S; i += 1024) sum += __expf(energies[i] - mx);
  red[tid] = sum;
  __syncthreads();
  for (int s = 512; s > 0; s >>= 1) {
    if (tid < s) red[tid] += red[tid + s];
    __syncthreads();
  }
  const float inv = 1.0f / red[0];

  for (int i = tid; i < S; i += 1024)
    out[i] = __expf(energies[i] - mx) * inv;
}

// ---------------------------------------------------------------------------
extern "C" void kernel_launch(void* const* d_in, const int* in_sizes, int n_in,
                              void* d_out, int out_size, void* d_ws, size_t ws_size,
                              hipStream_t stream) {
  const float* h   = (const float*)d_in[0];  // [1,1,H]
  const float* enc = (const float*)d_in[1];  // [1,S,H]
  const float* W   = (const float*)d_in[2];  // [H,H]
  const float* b   = (const float*)d_in[3];  // [H]
  float* out = (float*)d_out;                // [1,S]

  float* ws       = (float*)d_ws;            // [0..H-1]=v, [H]=c
  float* energies = ws + 2048;               // [S]

  attn_projvec_kernel<<<5, 256, 0, stream>>>(W, h, b, ws);
  attn_energies_kernel<<<S / ROWS, 256, 0, stream>>>(enc, ws, energies);
  attn_softmax_kernel<<<1, 1024, 0, stream>>>(energies, out);
}